// fofe_linear_tricontext_1374389534944
// MI455X (gfx1250) — compile-verified
//
#include <hip/hip_runtime.h>
#include <hip/hip_bf16.h>

typedef __attribute__((ext_vector_type(2))) float v2f;
typedef __attribute__((ext_vector_type(8))) float v8f;

#define DOC_LEN   46
#define N_CAND    415
#define EMBED     768
#define OUT_D     3840   // 5 * 768
#define BATCH_N   64
#define KSTEPS    12     // ceil(46/4) -> padded K = 48

// log2(0.9) for the alpha^t LDS table
#define LOG2_ALPHA (-0.15200309344504995f)

// span id -> (s, e)
__device__ __forceinline__ void span_of(int n, int& s, int& e) {
    if (n < 370) {            // i in [0,36]: 10 candidates each
        s = n / 10;
        e = s + (n % 10);
    } else {                  // triangle tail: sizes 9,8,...,1 for i=37..45
        int r = n - 370;
        int ss = 37, rem = 9;
        while (r >= rem) { r -= rem; --rem; ++ss; }
        s = ss;
        e = ss + r;
    }
}

// FOFE alpha buffer value for buffer q at (span s..e, position l). l may be >= DOC_LEN (pad).
__device__ __forceinline__ float buf_val(int q, int s, int e, int l, const float* pt) {
    if (l >= DOC_LEN) return 0.0f;
    switch (q) {
        case 0:  return (l >= s && l <= e) ? pt[e - l]     : 0.0f;  // cand
        case 1:  return (l < s)            ? pt[s - 1 - l] : 0.0f;  // left excl
        case 2:  return (l <= e)           ? pt[e - l]     : 0.0f;  // left incl
        case 3:  return (l > e)            ? pt[l - 1 - e] : 0.0f;  // right excl
        default: return (l >= s)           ? pt[l - s]     : 0.0f;  // right incl
    }
}

template <bool GUARDED>
__global__ void __launch_bounds__(256)
fofe_wmma_kernel(const float* __restrict__ x, float* __restrict__ out, int mtile0) {
    __shared__ float pow_tab[64];
    if (threadIdx.x < 64)
        pow_tab[threadIdx.x] = exp2f((float)threadIdx.x * LOG2_ALPHA);
    __syncthreads();

    const int mtile = mtile0 + blockIdx.x;  // 16 spans each
    const int b     = blockIdx.y;           // batch
    const int p     = blockIdx.z;           // output channel block 0..4
    // concat order [codes0, codes1, codes3, codes2, codes4]
    const int perm[5] = {0, 1, 3, 2, 4};
    const int q = perm[p];

    const int lane  = threadIdx.x & 31;
    const int wave  = threadIdx.x >> 5;     // 0..7
    const int half  = lane >> 4;            // 0 or 1
    const int lcol  = lane & 15;
    const int kbase = 2 * half;             // K offset inside the 4-wide step

    // ---- build the 12 A fragments for this wave's M-tile (shared across d-tiles)
    int n_row = mtile * 16 + lcol;          // span row this lane supplies for A
    int nclamp = n_row < N_CAND ? n_row : N_CAND - 1;  // garbage row masked at store
    int s, e;
    span_of(nclamp, s, e);

    v2f afrag[KSTEPS];
#pragma unroll
    for (int ks = 0; ks < KSTEPS; ++ks) {
        int l0 = ks * 4 + kbase;
        afrag[ks].x = buf_val(q, s, e, l0,     pow_tab);
        afrag[ks].y = buf_val(q, s, e, l0 + 1, pow_tab);
    }

    // ---- loop over this wave's 6 d-tiles
    const float* xb = x + (size_t)b * DOC_LEN * EMBED;
    const int nbase = mtile * 16 + 8 * half;   // first of the 8 rows this lane holds

#pragma unroll
    for (int i = 0; i < 6; ++i) {
        const int dt = wave * 6 + i;        // d-tile 0..47
        const int d  = dt * 16 + lcol;      // column inside EMBED
        v8f acc = {};
#pragma unroll
        for (int ks = 0; ks < KSTEPS; ++ks) {
            int l0 = ks * 4 + kbase;
            int l1 = l0 + 1;
            int l0c = l0 < DOC_LEN ? l0 : DOC_LEN - 1;   // clamp: stay in-bounds
            int l1c = l1 < DOC_LEN ? l1 : DOC_LEN - 1;
            v2f bv;
            bv.x = xb[l0c * EMBED + d] * (l0 < DOC_LEN ? 1.0f : 0.0f);
            bv.y = xb[l1c * EMBED + d] * (l1 < DOC_LEN ? 1.0f : 0.0f);
            // D = A(16x4 f32) * B(4x16 f32) + C
            acc = __builtin_amdgcn_wmma_f32_16x16x4_f32(
                /*neg_a=*/false, afrag[ks],
                /*neg_b=*/false, bv,
                /*c_mod=*/(short)0, acc,
                /*reuse_a=*/false, /*reuse_b=*/false);
        }
        // ---- store 16x16 tile: lane holds rows nbase..nbase+7, column dcol.
        // One base address per tile; the 8 row stores become immediate offsets
        // (r*OUT_D*4 <= 107520 B, fits the signed 24-bit IOFFSET).
        const int dcol = p * EMBED + dt * 16 + lcol;
        float* basep = out + ((size_t)b * N_CAND + nbase) * OUT_D + dcol;
        if (!GUARDED) {
#pragma unroll
            for (int r = 0; r < 8; ++r)
                __builtin_nontemporal_store(acc[r], basep + r * OUT_D);
        } else {
#pragma unroll
            for (int r = 0; r < 8; ++r)
                if (nbase + r < N_CAND)
                    __builtin_nontemporal_store(acc[r], basep + r * OUT_D);
        }
    }
}

__global__ void __launch_bounds__(256)
fofe_pos_kernel(float* __restrict__ pos) {
    int idx = blockIdx.x * blockDim.x + threadIdx.x;   // over BATCH * N_CAND
    if (idx >= BATCH_N * N_CAND) return;
    int n = idx % N_CAND;
    int s, e;
    span_of(n, s, e);
    pos[2 * idx]     = (float)s;
    pos[2 * idx + 1] = (float)e;
}

extern "C" void kernel_launch(void* const* d_in, const int* in_sizes, int n_in,
                              void* d_out, int out_size, void* d_ws, size_t ws_size,
                              hipStream_t stream) {
    (void)in_sizes; (void)n_in; (void)d_ws; (void)ws_size; (void)out_size;

    const float* x = (const float*)d_in[0];          // [64, 46, 768] f32
    float* out = (float*)d_out;                      // fofe_codes then positions
    float* pos = out + (size_t)BATCH_N * N_CAND * OUT_D;

    // mtiles 0..24: fully branchless store path (96% of the work)
    dim3 grid_fast(25, BATCH_N, 5);
    fofe_wmma_kernel<false><<<grid_fast, 256, 0, stream>>>(x, out, 0);

    // mtile 25: boundary tile (row n=415 masked)
    dim3 grid_edge(1, BATCH_N, 5);
    fofe_wmma_kernel<true><<<grid_edge, 256, 0, stream>>>(x, out, 25);

    int total = BATCH_N * N_CAND;
    fofe_pos_kernel<<<(total + 255) / 256, 256, 0, stream>>>(pos);
}